// OverflowDetectionHead_81587198755029
// MI455X (gfx1250) — compile-verified
//
#include <hip/hip_runtime.h>
#include <math.h>

typedef __attribute__((ext_vector_type(2))) float v2f;
typedef __attribute__((ext_vector_type(8))) float v8f;

// ---------------------------------------------------------------------------
// Kernel 0: zero the sums/counts workspace (harness poisons ws, atomics
// accumulate, so we must re-zero on every launch).
// ---------------------------------------------------------------------------
__global__ void zero_ws_kernel(float* __restrict__ ws, int n) {
    int i = blockIdx.x * blockDim.x + threadIdx.x;
    if (i < n) ws[i] = 0.0f;
}

// ---------------------------------------------------------------------------
// Kernel 1: segmented sum over sorted batch ids.
// Each lane loads feat[row][25], feat[row][9]; a wave32 segmented suffix-sum
// (Hillis-Steele with id guard, valid because ids are sorted -> runs are
// contiguous) leaves the full run-sum in each run-head lane; only heads do
// global atomicAdd. ~1-2 atomics per wave instead of 96.
// ---------------------------------------------------------------------------
__global__ __launch_bounds__(256)
void seg_reduce_kernel(const float* __restrict__ feat,
                       const int* __restrict__ batch,
                       float* __restrict__ sums,    // [ngraphs][2]
                       float* __restrict__ counts,  // [ngraphs]
                       int n, int nfeat) {
    int idx  = blockIdx.x * blockDim.x + threadIdx.x;
    int lane = threadIdx.x & 31;
    bool active = (idx < n);

    int   id = 0x7fffffff;            // sentinel: never equals a real id
    float v0 = 0.0f, v1 = 0.0f, vc = 0.0f;
    if (active) {
        id = batch[idx];
        const float* row = feat + (size_t)idx * (size_t)nfeat;
        v0 = row[25];
        v1 = row[9];
        vc = 1.0f;
    }

    // Segmented suffix sum within the wave (wave32).
    #pragma unroll
    for (int d = 1; d < 32; d <<= 1) {
        float o0 = __shfl_down(v0, d, 32);
        float o1 = __shfl_down(v1, d, 32);
        float oc = __shfl_down(vc, d, 32);
        int  oid = __shfl_down(id, d, 32);
        if ((lane + d) < 32 && oid == id) { v0 += o0; v1 += o1; vc += oc; }
    }

    int prev = __shfl_up(id, 1, 32);
    bool head = active && (lane == 0 || prev != id);
    if (head) {
        atomicAdd(&sums[(size_t)id * 2 + 0], v0);
        atomicAdd(&sums[(size_t)id * 2 + 1], v1);
        atomicAdd(&counts[id], vc);
    }
}

// ---------------------------------------------------------------------------
// Kernel 2: per-graph MLP head via V_WMMA_F32_16X16X4_F32.
// One wave (32 threads) handles 16 graphs.
//   A (16x4 f32): lanes 0-15 hold {risk0, risk1} (K=0,1); lanes 16-31 hold
//                 K=2,3 which are zero-padding.
//   B (4x16 f32): lanes 0-15 hold {W1[0][col], W1[1][col]}; lanes 16-31 zero.
//   C init = b1[col] broadcast (C[m][n] = b1[n]).
// Four N-tiles cover the 64 hidden units. Layer 2 (x W2, relu, 64->1) is a
// per-lane weighted sum over the C tile + 16-lane shfl_xor butterfly.
// ---------------------------------------------------------------------------
__global__ __launch_bounds__(32)
void mlp_head_kernel(const float* __restrict__ sums,
                     const float* __restrict__ counts,
                     const float* __restrict__ W1,   // [2][64]
                     const float* __restrict__ b1,   // [64]
                     const float* __restrict__ W2,   // [64][1]
                     const float* __restrict__ b2,   // [1]
                     float* __restrict__ out, int ngraphs) {
    int gbase = blockIdx.x * 16;
    int lane  = threadIdx.x;      // 0..31
    int col   = lane & 15;        // N index within the 16x16 C tile

    // Build the A fragment (risk matrix rows, K padded 2->4 with zeros).
    v2f A = {0.0f, 0.0f};
    if (lane < 16) {
        int g = gbase + lane;
        if (g < ngraphs) {
            float c  = counts[g];
            float cm = fmaxf(c, 1.0f);
            float m0 = sums[(size_t)g * 2 + 0] / cm;
            float m1 = sums[(size_t)g * 2 + 1] / cm;
            A.x = 1.0f - m0;   // risk0
            A.y = m1;          // risk1
        }
    }

    float acc[8];
    #pragma unroll
    for (int r = 0; r < 8; ++r) acc[r] = 0.0f;

    #pragma unroll
    for (int t = 0; t < 4; ++t) {
        // B fragment: K=0,1 rows of W1 in lanes 0-15; K=2,3 (lanes 16-31) zero.
        v2f B = {0.0f, 0.0f};
        if (lane < 16) {
            B.x = W1[t * 16 + lane];        // W1[0][t*16+n]
            B.y = W1[64 + t * 16 + lane];   // W1[1][t*16+n]
        }
        float bias = b1[t * 16 + col];
        v8f C = {bias, bias, bias, bias, bias, bias, bias, bias};

        // D = A x B + C  (16x16 f32 accumulator, full precision)
        C = __builtin_amdgcn_wmma_f32_16x16x4_f32(
                /*neg_a=*/false, A, /*neg_b=*/false, B,
                /*c_mod=*/(short)0, C, /*reuse_a=*/false, /*reuse_b=*/false);

        // Layer 2 partial: relu(h) * W2 for this lane's column.
        float w2 = W2[t * 16 + col];
        #pragma unroll
        for (int r = 0; r < 8; ++r)
            acc[r] += fmaxf(C[r], 0.0f) * w2;
    }

    // Reduce across the 16 lanes sharing the same M-half (xor<=8 stays
    // within each group of 16).
    #pragma unroll
    for (int off = 8; off >= 1; off >>= 1) {
        #pragma unroll
        for (int r = 0; r < 8; ++r)
            acc[r] += __shfl_xor(acc[r], off, 32);
    }

    // Lane 0 writes graphs gbase+0..7 (M=0..7), lane 16 writes gbase+8..15.
    if (col == 0) {
        int mbase = (lane >= 16) ? 8 : 0;
        float bb = b2[0];
        #pragma unroll
        for (int r = 0; r < 8; ++r) {
            int g = gbase + mbase + r;
            if (g < ngraphs) {
                float s = 1.0f / (1.0f + expf(-(acc[r] + bb)));
                out[g] = (counts[g] > 0.0f) ? s : 0.0f;
            }
        }
    }
}

// ---------------------------------------------------------------------------
// Launch
// ---------------------------------------------------------------------------
extern "C" void kernel_launch(void* const* d_in, const int* in_sizes, int n_in,
                              void* d_out, int out_size, void* d_ws, size_t ws_size,
                              hipStream_t stream) {
    const float* feat  = (const float*)d_in[0];  // node_features [N][64]
    const int*   batch = (const int*)d_in[1];    // [N], sorted
    // d_in[2] = graph_embedding (unused by the reference)
    const float* W1 = (const float*)d_in[3];     // [2][64]
    const float* b1 = (const float*)d_in[4];     // [64]
    const float* W2 = (const float*)d_in[5];     // [64][1]
    const float* b2 = (const float*)d_in[6];     // [1]
    float* out = (float*)d_out;

    int n       = in_sizes[1];
    int nfeat   = in_sizes[0] / n;
    int ngraphs = out_size;

    float* sums   = (float*)d_ws;                    // [ngraphs][2]
    float* counts = sums + (size_t)ngraphs * 2;      // [ngraphs]

    int zn = ngraphs * 3;
    zero_ws_kernel<<<(zn + 255) / 256, 256, 0, stream>>>(sums, zn);
    seg_reduce_kernel<<<(n + 255) / 256, 256, 0, stream>>>(feat, batch, sums, counts, n, nfeat);
    mlp_head_kernel<<<(ngraphs + 15) / 16, 32, 0, stream>>>(sums, counts, W1, b1, W2, b2, out, ngraphs);
}